// Nagasaki_40458591928608
// MI455X (gfx1250) — compile-verified
//
#include <hip/hip_runtime.h>
#include <hip/hip_bf16.h>

// ---------------------------------------------------------------------------
// Graph-transformer attention with adjacency-power bias, CDNA5 (gfx1250).
// All dense matmuls run on the matrix cores via V_WMMA_F32_16X16X4_F32
// (f32 in / f32 out, matches reference precision; P^5 chains 4 matmuls so
// bf16 inputs would compound too much error).
// ---------------------------------------------------------------------------

typedef __attribute__((ext_vector_type(2))) float v2f;
typedef __attribute__((ext_vector_type(8))) float v8f;

#define B_  16
#define N_  512
#define D_  512
#define H_  8
#define HD_ 64
#define FH_ 12

#define PA  68     // A tile pitch (floats): 68 mod 64 = 4 -> perfect bank spread
#define PB  162    // B pair-row pitch (floats): reads ~conflict-free, 2*PB%64=4

// ---------------------------------------------------------------------------
// Generic batched GEMM:  C = alpha * A(@)B  [+ C if beta] [+ biasVec[col]]
//   A: [M,K] row-major, row stride lda,  per-batch offset z -> (b,h) split
//   B: [K,N] row-major (transB=0) or [N,K] row-major (transB=1)
//   block = 128 threads = 4 waves; block tile 64x64; wave tile 32x32
//   K-step 4 per V_WMMA_F32_16X16X4_F32.
// Fragment layouts per CDNA5 ISA 7.12.2:
//   A 16x4: lane&15 = M row, (lane>>4)*2 = K base, 2 VGPRs = K, K+1
//   B 4x16: lane&15 = N col, (lane>>4)*2 = K base (mirrors A)
//   C 16x16: lane&15 = N col, vgpr r -> M = r + (lane>>4)*8
// LDS: A stored row-major (pitch PA); B stored k-paired: element (k,n) at
//   lB[(k>>1)*PB + 2n + (k&1)] so each lane's B fragment is ONE b64 load.
// ---------------------------------------------------------------------------
__global__ void __launch_bounds__(128)
k_wmma_gemm(const float* __restrict__ A, const float* __restrict__ Bm,
            float* __restrict__ C, const float* __restrict__ biasVec,
            int lda, int ldb, int ldc, int K,
            long sAb, long sAh, long sBb, long sBh, long sCb, long sCh,
            int nH, int transB, float alpha, int beta)
{
    const int z  = blockIdx.z;
    const int bb = z / nH, hh = z % nH;
    const float* __restrict__ Ab = A  + (long)bb * sAb + (long)hh * sAh;
    const float* __restrict__ Bb = Bm + (long)bb * sBb + (long)hh * sBh;
    float* __restrict__       Cb = C  + (long)bb * sCb + (long)hh * sCh;

    const int m0 = blockIdx.x * 64;
    const int n0 = blockIdx.y * 64;

    __shared__ __align__(16) float lA[64 * PA];   // 17 KB
    __shared__ __align__(16) float lB[32 * PB];   // 20.25 KB

    const int t    = threadIdx.x;
    const int lane = t & 31;
    const int wv   = t >> 5;
    const int wm   = (wv >> 1) * 32;   // wave M offset in tile
    const int wn   = (wv & 1)  * 32;   // wave N offset in tile

    v8f acc[2][2] = {};

    for (int k0 = 0; k0 < K; k0 += 64) {
        // ---- prefetch next k-tile into cache (global_prefetch_b8) ----
        if (k0 + 64 < K) {
            int pr = t >> 1, pc = (t & 1) * 32;
            __builtin_prefetch(Ab + (long)(m0 + pr) * lda + (k0 + 64 + pc), 0, 0);
            __builtin_prefetch(transB
                ? (Bb + (long)(n0 + pr) * ldb + (k0 + 64 + pc))
                : (Bb + (long)(k0 + 64 + pr) * ldb + (n0 + pc)), 0, 0);
        }
        // ---- load A tile 64x64 (16 float4 per row, 8 float4 per thread) ----
        #pragma unroll
        for (int i = 0; i < 8; ++i) {
            int f  = t + i * 128;          // 0..1023
            int r  = f >> 4;
            int c4 = (f & 15) * 4;
            float4 v = *(const float4*)(Ab + (long)(m0 + r) * lda + (k0 + c4));
            *(float4*)&lA[r * PA + c4] = v;
        }
        // ---- load B tile into k-paired layout ----
        if (!transB) {
            #pragma unroll
            for (int i = 0; i < 8; ++i) {
                int f  = t + i * 128;
                int r  = f >> 4;           // k index
                int c4 = (f & 15) * 4;     // n index
                float4 v = *(const float4*)(Bb + (long)(k0 + r) * ldb + (n0 + c4));
                float* d = &lB[(r >> 1) * PB + (r & 1)];
                d[(c4 + 0) * 2] = v.x;
                d[(c4 + 1) * 2] = v.y;
                d[(c4 + 2) * 2] = v.z;
                d[(c4 + 3) * 2] = v.w;
            }
        } else {
            // B stored [N,K]; element (k = c4+j, n = r)
            #pragma unroll
            for (int i = 0; i < 8; ++i) {
                int f  = t + i * 128;
                int r  = f >> 4;           // n index
                int c4 = (f & 15) * 4;     // k index
                float4 v = *(const float4*)(Bb + (long)(n0 + r) * ldb + (k0 + c4));
                lB[((c4 + 0) >> 1) * PB + r * 2 + ((c4 + 0) & 1)] = v.x;
                lB[((c4 + 1) >> 1) * PB + r * 2 + ((c4 + 1) & 1)] = v.y;
                lB[((c4 + 2) >> 1) * PB + r * 2 + ((c4 + 2) & 1)] = v.z;
                lB[((c4 + 3) >> 1) * PB + r * 2 + ((c4 + 3) & 1)] = v.w;
            }
        }
        __syncthreads();

        const int row = lane & 15;
        const int kg  = (lane >> 4) << 1;          // 0 or 2
        #pragma unroll
        for (int kk = 0; kk < 64; kk += 4) {
            const int p = (kk + kg) >> 1;          // k-pair index
            v2f a0 = *(const v2f*)&lA[(wm + row)      * PA + kk + kg];
            v2f a1 = *(const v2f*)&lA[(wm + 16 + row) * PA + kk + kg];
            v2f b0 = *(const v2f*)&lB[p * PB + (wn + row) * 2];
            v2f b1 = *(const v2f*)&lB[p * PB + (wn + 16 + row) * 2];

            acc[0][0] = __builtin_amdgcn_wmma_f32_16x16x4_f32(
                false, a0, false, b0, (short)0, acc[0][0], false, false);
            acc[0][1] = __builtin_amdgcn_wmma_f32_16x16x4_f32(
                false, a0, false, b1, (short)0, acc[0][1], false, false);
            acc[1][0] = __builtin_amdgcn_wmma_f32_16x16x4_f32(
                false, a1, false, b0, (short)0, acc[1][0], false, false);
            acc[1][1] = __builtin_amdgcn_wmma_f32_16x16x4_f32(
                false, a1, false, b1, (short)0, acc[1][1], false, false);
        }
        __syncthreads();
    }

    // ---- epilogue ----
    const int col_l = lane & 15;
    const int rbase = (lane >> 4) * 8;
    #pragma unroll
    for (int mi = 0; mi < 2; ++mi) {
        #pragma unroll
        for (int ni = 0; ni < 2; ++ni) {
            int row0 = m0 + wm + mi * 16 + rbase;
            int col  = n0 + wn + ni * 16 + col_l;
            float bval = biasVec ? biasVec[col] : 0.0f;
            #pragma unroll
            for (int r = 0; r < 8; ++r) {
                long idx = (long)(row0 + r) * ldc + col;
                float v = alpha * acc[mi][ni][r] + bval;
                if (beta) v += Cb[idx];
                Cb[idx] = v;
            }
        }
    }
}

// ---------------------------------------------------------------------------
// P = rownorm(mask ? adj : 0);  one block per (b,i) row, 256 threads x 2 cols
// ---------------------------------------------------------------------------
__global__ void __launch_bounds__(256)
k_rownorm(const float* __restrict__ adj, const int* __restrict__ mask,
          float* __restrict__ P)
{
    const int row = blockIdx.x;          // 0 .. B*N-1
    const int b = row >> 9;
    const int i = row & (N_ - 1);
    const int t = threadIdx.x;

    const int   mi   = mask[b * N_ + i];
    const float* ar  = adj + (long)row * N_;
    float*       pr  = P   + (long)row * N_;

    float v0 = 0.f, v1 = 0.f;
    if (mi) {
        v0 = mask[b * N_ + t]        ? ar[t]        : 0.f;
        v1 = mask[b * N_ + t + 256]  ? ar[t + 256]  : 0.f;
    }
    __shared__ float red[256];
    red[t] = v0 + v1;
    __syncthreads();
    #pragma unroll
    for (int o = 128; o > 0; o >>= 1) {
        if (t < o) red[t] += red[t + o];
        __syncthreads();
    }
    float rs  = red[0];
    float inv = (rs == 0.f) ? 1.f : 1.f / rs;
    pr[t]       = v0 * inv;
    pr[t + 256] = v1 * inv;
}

// ---------------------------------------------------------------------------
// Edge FFN: e = [I, P, P^2..P^5](b,i,j) -> relu(e@w1+b1)@w2+b2 -> bias[B,H,N,N]
// ---------------------------------------------------------------------------
__global__ void __launch_bounds__(256)
k_edge_ffn(const float* __restrict__ P1, const float* __restrict__ PW,
           const float* __restrict__ w1, const float* __restrict__ b1,
           const float* __restrict__ w2, const float* __restrict__ b2,
           float* __restrict__ bias)
{
    const long BNN = (long)B_ * N_ * N_;
    long idx = (long)blockIdx.x * 256 + threadIdx.x;   // < B*N*N
    int  b = (int)(idx >> 18);                         // /(N*N)
    long r = idx & ((long)N_ * N_ - 1);
    int  i = (int)(r >> 9);
    int  j = (int)(r & (N_ - 1));

    float e[6];
    e[0] = (i == j) ? 1.f : 0.f;
    e[1] = P1[idx];
    e[2] = PW[idx];
    e[3] = PW[idx + BNN];
    e[4] = PW[idx + 2 * BNN];
    e[5] = PW[idx + 3 * BNN];

    float o[H_];
    #pragma unroll
    for (int hh = 0; hh < H_; ++hh) o[hh] = b2[hh];

    #pragma unroll
    for (int tt = 0; tt < FH_; ++tt) {
        float hs = b1[tt];
        #pragma unroll
        for (int s = 0; s < 6; ++s) hs += e[s] * w1[s * FH_ + tt];
        hs = fmaxf(hs, 0.f);
        #pragma unroll
        for (int hh = 0; hh < H_; ++hh) o[hh] += hs * w2[tt * H_ + hh];
    }
    const long base = ((long)b * H_ * N_ + (long)i) * N_ + j;
    #pragma unroll
    for (int hh = 0; hh < H_; ++hh)
        bias[base + (long)hh * N_ * N_] = o[hh];
}

// ---------------------------------------------------------------------------
// Masked softmax in place over last dim (512); one block per (b,h,q) row.
// ---------------------------------------------------------------------------
__global__ void __launch_bounds__(256)
k_softmax(float* __restrict__ attn, const int* __restrict__ mask)
{
    const long row = blockIdx.x;                 // 0 .. B*H*N-1
    const int  b   = (int)(row / ((long)H_ * N_));
    float* p = attn + row * N_;
    const int* mrow = mask + (long)b * N_;
    const int t = threadIdx.x;

    float x0 = mrow[t]       ? p[t]       : -1e9f;
    float x1 = mrow[t + 256] ? p[t + 256] : -1e9f;

    __shared__ float red[256];
    red[t] = fmaxf(x0, x1);
    __syncthreads();
    #pragma unroll
    for (int o = 128; o > 0; o >>= 1) {
        if (t < o) red[t] = fmaxf(red[t], red[t + o]);
        __syncthreads();
    }
    float mx = red[0];
    __syncthreads();

    float e0 = __expf(x0 - mx);
    float e1 = __expf(x1 - mx);
    red[t] = e0 + e1;
    __syncthreads();
    #pragma unroll
    for (int o = 128; o > 0; o >>= 1) {
        if (t < o) red[t] += red[t + o];
        __syncthreads();
    }
    float inv = 1.f / red[0];
    p[t]       = e0 * inv;
    p[t + 256] = e1 * inv;
}

// ---------------------------------------------------------------------------
// atten_weights = mean over heads of attn
// ---------------------------------------------------------------------------
__global__ void __launch_bounds__(256)
k_head_mean(const float* __restrict__ attn, float* __restrict__ outW)
{
    const long NN = (long)N_ * N_;
    long idx = (long)blockIdx.x * 256 + threadIdx.x;  // < B*N*N
    int  b = (int)(idx >> 18);
    long r = idx & (NN - 1);
    const float* p = attn + (long)b * H_ * NN + r;
    float s = 0.f;
    #pragma unroll
    for (int hh = 0; hh < H_; ++hh) s += p[hh * NN];
    outW[idx] = s * (1.0f / H_);
}

// ---------------------------------------------------------------------------
extern "C" void kernel_launch(void* const* d_in, const int* in_sizes, int n_in,
                              void* d_out, int out_size, void* d_ws, size_t ws_size,
                              hipStream_t stream)
{
    (void)in_sizes; (void)n_in; (void)out_size; (void)ws_size;
    const float* h     = (const float*)d_in[0];
    const float* adj   = (const float*)d_in[1];
    const int*   mask  = (const int*)  d_in[2];
    const float* in_w  = (const float*)d_in[3];
    const float* in_b  = (const float*)d_in[4];
    const float* out_w = (const float*)d_in[5];
    const float* out_b = (const float*)d_in[6];
    const float* w1    = (const float*)d_in[7];
    const float* b1    = (const float*)d_in[8];
    const float* w2    = (const float*)d_in[9];
    const float* b2    = (const float*)d_in[10];

    float* out       = (float*)d_out;
    float* atten_out = out;                          // [B,N,D]
    float* atten_w   = out + (long)B_ * N_ * D_;     // [B,N,N]

    const long NN  = (long)N_ * N_;
    const long BNN = (long)B_ * NN;
    float* ws   = (float*)d_ws;
    float* P    = ws;                 // 1*BNN : transition matrix
    float* PW   = P + BNN;            // 4*BNN : P^2..P^5 (later reused as qkv)
    float* BIAS = PW + 4 * BNN;       // 8*BNN : bias -> scores -> attn
    float* qkv  = PW;                 // [B,N,3D] (powers dead after edge FFN)
    float* ctx  = P;                  // [B,N,D]  (P dead after edge FFN)

    auto gemm = [&](const float* A, const float* Bm, float* C, const float* bv,
                    int lda, int ldb, int ldc, int K,
                    long sAb, long sAh, long sBb, long sBh, long sCb, long sCh,
                    int nH, int transB, float alpha, int beta,
                    int tM, int tN, int nz) {
        k_wmma_gemm<<<dim3(tM, tN, nz), 128, 0, stream>>>(
            A, Bm, C, bv, lda, ldb, ldc, K,
            sAb, sAh, sBb, sBh, sCb, sCh, nH, transB, alpha, beta);
    };

    // 1) P = rownorm(masked adj)
    k_rownorm<<<dim3(B_ * N_), 256, 0, stream>>>(adj, mask, P);

    // 2) matrix powers P^2..P^5 (batched over B)
    gemm(P,            P, PW,            nullptr, N_, N_, N_, N_,
         NN, 0, NN, 0, NN, 0, 1, 0, 1.f, 0, 8, 8, B_);
    gemm(PW,           P, PW + BNN,      nullptr, N_, N_, N_, N_,
         NN, 0, NN, 0, NN, 0, 1, 0, 1.f, 0, 8, 8, B_);
    gemm(PW + BNN,     P, PW + 2 * BNN,  nullptr, N_, N_, N_, N_,
         NN, 0, NN, 0, NN, 0, 1, 0, 1.f, 0, 8, 8, B_);
    gemm(PW + 2 * BNN, P, PW + 3 * BNN,  nullptr, N_, N_, N_, N_,
         NN, 0, NN, 0, NN, 0, 1, 0, 1.f, 0, 8, 8, B_);

    // 3) per-pair edge FFN -> BIAS [B,H,N,N]
    k_edge_ffn<<<dim3((unsigned)(BNN / 256)), 256, 0, stream>>>(
        P, PW, w1, b1, w2, b2, BIAS);

    // 4) qkv = h @ in_proj_w^T + in_proj_b   [B,N,3D]   (reuses PW storage)
    gemm(h, in_w, qkv, in_b, D_, D_, 3 * D_, D_,
         (long)N_ * D_, 0, 0, 0, (long)N_ * 3 * D_, 0,
         1, 1, 1.f, 0, 8, 24, B_);

    // 5) scores = 0.125 * Q K^T + BIAS  (in place in BIAS, beta=1), z=(b,h)
    gemm(qkv, qkv + D_, BIAS, nullptr, 3 * D_, 3 * D_, N_, HD_,
         (long)N_ * 3 * D_, HD_, (long)N_ * 3 * D_, HD_,
         (long)H_ * NN, NN, H_, 1, 0.125f, 1, 8, 8, B_ * H_);

    // 6) masked softmax in place
    k_softmax<<<dim3(B_ * H_ * N_), 256, 0, stream>>>(BIAS, mask);

    // 7) ctx = attn @ V   [B,N,D] interleaved by head column blocks
    gemm(BIAS, qkv + 2 * D_, ctx, nullptr, N_, 3 * D_, D_, N_,
         (long)H_ * NN, NN, (long)N_ * 3 * D_, HD_,
         (long)N_ * D_, HD_, H_, 0, 1.f, 0, 8, 1, B_ * H_);

    // 8) atten_out = ctx @ out_w^T + out_b
    gemm(ctx, out_w, atten_out, out_b, D_, D_, D_, D_,
         (long)N_ * D_, 0, 0, 0, (long)N_ * D_, 0,
         1, 1, 1.f, 0, 8, 8, B_);

    // 9) atten_weights = head mean of attn
    k_head_mean<<<dim3((unsigned)(BNN / 256)), 256, 0, stream>>>(BIAS, atten_w);
}